// NeighborhoodShift2d_46110768890403
// MI455X (gfx1250) — compile-verified
//
#include <hip/hip_runtime.h>

// NATTEN 7x7 neighborhood shift, dilation 1, zero padding.
// x:   [1, 8, 32, 128, 128]  f32   (16 MiB  -> L2 resident)
// out: [1, 8, 49, 32, 128, 128] f32 (784 MiB -> pure streaming writes, HBM bound)
//
// Strategy: per-(head, chan, 16-row tile) workgroup stages input rows (+zero
// halo) into LDS via global_load_async_to_lds_b128, then emits all 49 shifted
// views with global_store_async_from_lds_b32 (ASYNCcnt DMA path, NT stores).

#define RAD        3
#define TILE_Y     16
#define LDS_ROWS   (TILE_Y + 2 * RAD)        // 22 input rows per tile
#define HALO_X     4                          // 4-float halo keeps b128 LDS alignment
#define LDS_STRIDE (HALO_X + 128 + HALO_X)    // 136 floats per LDS row
#define LDS_FLOATS (LDS_ROWS * LDS_STRIDE)    // 2992 floats = 11968 B

#define CPOL_STORE_NT 1                       // TH[2:0]=1 -> non-temporal store

typedef float v4f   __attribute__((ext_vector_type(4)));
typedef int   i32x4 __attribute__((vector_size(16)));   // matches builtin param type

#define GAS __attribute__((address_space(1)))
#define LAS __attribute__((address_space(3)))

#if defined(__has_builtin)
#  if __has_builtin(__builtin_amdgcn_global_load_async_to_lds_b128)
#    define HAS_ASYNC_LOAD 1
#  endif
#  if __has_builtin(__builtin_amdgcn_global_store_async_from_lds_b32)
#    define HAS_ASYNC_STORE 1
#  endif
#  if __has_builtin(__builtin_amdgcn_s_wait_asynccnt)
#    define HAS_WAIT_ASYNC 1
#  endif
#endif
#ifndef HAS_ASYNC_LOAD
#  define HAS_ASYNC_LOAD 0
#endif
#ifndef HAS_ASYNC_STORE
#  define HAS_ASYNC_STORE 0
#endif
#ifndef HAS_WAIT_ASYNC
#  define HAS_WAIT_ASYNC 0
#endif

__device__ __forceinline__ void wait_async_zero() {
#if HAS_WAIT_ASYNC
  __builtin_amdgcn_s_wait_asynccnt(0);
#else
  asm volatile("s_wait_asynccnt 0" ::: "memory");
#endif
}

__global__ __launch_bounds__(256, 1)
void natten49_shift_kernel(const float* __restrict__ x, float* __restrict__ out) {
  __shared__ float lds[LDS_FLOATS];

  const int tid = threadIdx.x;
  const int bid = blockIdx.x;      // 2048 blocks = 8 heads * 32 chans * 8 y-tiles
  const int ty  = bid & 7;         // y-tile (16 rows each)
  const int c   = (bid >> 3) & 31; // channel
  const int h   = bid >> 8;        // head
  const int y0  = ty * TILE_Y;

  const float* __restrict__ plane = x + (((size_t)(h * 32 + c)) << 14); // 128*128

  // ---- 1) zero-fill LDS tile (zero halo == free boundary padding) ----
  {
    v4f z = {0.f, 0.f, 0.f, 0.f};
    v4f* l4 = (v4f*)lds;
    for (int i = tid; i < LDS_FLOATS / 4; i += 256)
      l4[i] = z;
  }
  __syncthreads();

  // ---- 2) stage input rows [y0-3, y0+18] into LDS (async b128 DMA) ----
  for (int m = tid; m < LDS_ROWS * 32; m += 256) {
    const int r   = m >> 5;          // LDS row 0..21
    const int q4  = (m & 31) << 2;   // float4 column * 4
    const int yin = y0 - RAD + r;    // input row
    if (0 <= yin && yin < 128) {
      const float* src = plane + yin * 128 + q4;
      float*       dst = lds + r * LDS_STRIDE + HALO_X + q4; // 16B aligned
#if HAS_ASYNC_LOAD
      __builtin_amdgcn_global_load_async_to_lds_b128(
          (GAS i32x4*)src, (LAS i32x4*)dst, 0, 0);
#else
      *(v4f*)dst = *(const v4f*)src;
#endif
    }
  }
#if HAS_ASYNC_LOAD
  wait_async_zero();
#endif
  __syncthreads();

  // ---- 3) emit the 49 shifted views of this tile ----
  // out index: (((h*49 + k)*32 + c)*128 + y)*128 + x
  const size_t base = (((size_t)(h * (49 * 32) + c)) << 14) + (size_t)y0 * 128;

  int k = 0;
  for (int dy = -RAD; dy <= RAD; ++dy) {
    for (int dx = -RAD; dx <= RAD; ++dx, ++k) {
      float* __restrict__ ob = out + base + (size_t)k * (size_t)(32 << 14);
      const float* lrow = lds + (RAD + dy) * LDS_STRIDE + (HALO_X + dx);
#if HAS_ASYNC_STORE
      // Entire tile written by the async LDS->global engine (ASYNCcnt).
      // Lanes write consecutive 4B addrs -> 128B/wave fully coalesced;
      // LDS reads are stride-1 floats -> bank-conflict free. NT temporal
      // hint: output is write-once streaming data, never re-read.
#pragma unroll
      for (int it = 0; it < (TILE_Y * 128) / 256; ++it) {   // 8 per k
        const int m  = tid + it * 256;    // 0..2047
        const int yl = m >> 7;            // 0..15
        const int xx = m & 127;           // 0..127
        __builtin_amdgcn_global_store_async_from_lds_b32(
            (GAS int*)(ob + yl * 128 + xx),
            (LAS int*)(lrow + yl * LDS_STRIDE + xx), 0, CPOL_STORE_NT);
      }
#else
      // Fallback: LDS gather + non-temporal b128 stores (512B/wave).
#pragma unroll
      for (int j = 0; j < (TILE_Y * 32) / 256; ++j) {       // 2 per k
        const int i  = tid + j * 256;     // 0..511
        const int yl = i >> 5;            // 0..15
        const int xx = (i & 31) << 2;     // 0,4,...,124
        const float* s = lrow + yl * LDS_STRIDE + xx;
        v4f v;
        v.x = s[0]; v.y = s[1]; v.z = s[2]; v.w = s[3];
        __builtin_nontemporal_store(v, (v4f*)(ob + yl * 128 + xx));
      }
#endif
    }
  }
  // s_endpgm performs an implicit wait-idle, draining outstanding async stores.
}

extern "C" void kernel_launch(void* const* d_in, const int* in_sizes, int n_in,
                              void* d_out, int out_size, void* d_ws, size_t ws_size,
                              hipStream_t stream) {
  (void)in_sizes; (void)n_in; (void)d_ws; (void)ws_size; (void)out_size;
  const float* x  = (const float*)d_in[0];
  float* out      = (float*)d_out;
  // 8 heads * 32 channels * 8 y-tiles = 2048 workgroups, 256 threads (8 wave32)
  natten49_shift_kernel<<<dim3(2048), dim3(256), 0, stream>>>(x, out);
}